// vaeVectorQuantizer_23974507446775
// MI455X (gfx1250) — compile-verified
//
#include <hip/hip_runtime.h>

typedef __attribute__((ext_vector_type(2))) float v2f;
typedef __attribute__((ext_vector_type(4))) float v4f;
typedef __attribute__((ext_vector_type(8))) float v8f;

#define DDIM 64
#define KCODES 1024
#define ROWS_PER_WAVE 16
#define WAVES_PER_WG 8
#define ROWS_PER_WG (ROWS_PER_WAVE * WAVES_PER_WG)   // 128
#define THREADS 256
#define NTILES (KCODES / 16)                          // 64 column tiles

// LDS layout (dynamic):
//   ldsA   : codebook pre-swizzled into WMMA A-fragment order
//            [tile ct][lane][16 x v2f] = 64*32*32 floats = 256 KB
//   ldsEsq : -0.5 * ||e_k||^2, 1024 f32
//   ldsCode: 128 i32
#define LDS_E_ELEMS (DDIM * KCODES)                   // 65536 floats
#define LDS_BYTES ((LDS_E_ELEMS + KCODES) * 4 + ROWS_PER_WG * 4)

#if defined(__gfx1250__) && \
    __has_builtin(__builtin_amdgcn_global_load_async_to_lds_b32) && \
    __has_builtin(__builtin_amdgcn_global_store_async_from_lds_b32) && \
    __has_builtin(__builtin_amdgcn_s_wait_asynccnt)
#define USE_ASYNC_LDS 1
typedef __attribute__((address_space(1))) int* gint_t;   // global int*
typedef __attribute__((address_space(3))) int* lint_t;   // LDS int*
#else
#define USE_ASYNC_LDS 0
#endif

__global__ __launch_bounds__(THREADS)
void vq_argmin_gather_kernel(const float* __restrict__ x,
                             const float* __restrict__ emb,   // [D][K] row-major
                             float* __restrict__ out)
{
    extern __shared__ float lds[];
    float* ldsA    = lds;                                // swizzled codebook
    float* ldsEsq  = lds + LDS_E_ELEMS;                  // [K], holds -0.5*e_sq
    int*   ldsCode = (int*)(lds + LDS_E_ELEMS + KCODES); // [128]

    const int tid  = threadIdx.x;
    const int wave = tid >> 5;
    const int lane = tid & 31;
    const int lo   = lane & 15;
    const int hi   = lane >> 4;

    // ---- Stage codebook into fragment-swizzled LDS layout.
    // Element emb[d][k] -> ldsA[(ct*32 + lane')*32 + j*2 + pos]
    //   ct = k>>4, lane' = (k&15) + 16*((d&3)>>1), j = d>>2, pos = d&1
#if USE_ASYNC_LDS
    {
        // Per-lane global gather -> per-lane LDS scatter on the async copy
        // engine; no VGPR round trip. 128 B per wave-instruction.
        #pragma unroll 4
        for (int i = 0; i < LDS_E_ELEMS / THREADS; ++i) {   // 256 iters
            const int e  = tid + i * THREADS;   // coalesced per wave
            const int d  = e >> 10;
            const int k  = e & (KCODES - 1);
            const int ct = k >> 4;
            const int l2 = (k & 15) + 16 * ((d & 3) >> 1);
            float* gsrc = const_cast<float*>(emb + e);
            float* dst  = ldsA + (ct * 32 + l2) * 32 + (d >> 2) * 2 + (d & 1);
            __builtin_amdgcn_global_load_async_to_lds_b32(
                (gint_t)gsrc, (lint_t)dst, 0, 0);
        }
        __builtin_amdgcn_s_wait_asynccnt(0);
    }
#else
    {
        const v4f* __restrict__ embv = (const v4f*)emb;
        #pragma unroll
        for (int i = 0; i < (LDS_E_ELEMS / 4) / THREADS; ++i) {
            const int e4 = tid + i * THREADS;      // v4f index; 256 v4f per d-row
            const v4f v  = embv[e4];
            const int d  = e4 >> 8;
            const int k0 = (e4 & 255) * 4;
            const int j   = d >> 2;
            const int rem = d & 3;
            const int hh  = rem >> 1;
            const int pos = rem & 1;
            #pragma unroll
            for (int c = 0; c < 4; ++c) {
                const int k  = k0 + c;
                const int ct = k >> 4;
                const int l2 = (k & 15) + 16 * hh;
                ldsA[(ct * 32 + l2) * 32 + j * 2 + pos] = v[c];
            }
        }
    }
#endif
    __syncthreads();

    // ---- ldsEsq[k] = -0.5 * sum_d e^2 (read back from swizzled blocks) ----
    #pragma unroll
    for (int i = 0; i < KCODES / THREADS; ++i) {
        const int k  = tid + i * THREADS;
        const int ct = k >> 4;
        const int l2 = k & 15;
        const v4f* b0 = (const v4f*)(ldsA + (ct * 32 + l2) * 32);      // d%4 in {0,1}
        const v4f* b1 = (const v4f*)(ldsA + (ct * 32 + l2 + 16) * 32); // d%4 in {2,3}
        float s = 0.f;
        #pragma unroll
        for (int q = 0; q < 8; ++q) {
            const v4f u = b0[q], w = b1[q];
            s += u.x * u.x + u.y * u.y + u.z * u.z + u.w * u.w;
            s += w.x * w.x + w.y * w.y + w.z * w.z + w.w * w.w;
        }
        ldsEsq[k] = -0.5f * s;
    }
    __syncthreads();

    // ---- X^T B-fragments in registers (16 k-steps of K=4) ----
    const int rowbase = blockIdx.x * ROWS_PER_WG + wave * ROWS_PER_WAVE;
    const int row = rowbase + lo;
    v2f bfrag[DDIM / 4];
    #pragma unroll
    for (int j = 0; j < DDIM / 4; ++j) {
        const int d0 = 4 * j + 2 * hi;
        bfrag[j] = *(const v2f*)(x + (size_t)row * DDIM + d0);
    }

    // ---- Main loop: acc starts at -0.5*||e||^2, ends at sim - 0.5*||e||^2.
    //      argmin(dist) == argmax(acc). ----
    float best    = -3.402823e38f;
    int   bestIdx = 0;

    for (int ct = 0; ct < NTILES; ++ct) {
        const int cb = ct * 16;
        const float* ablk = ldsA + (ct * 32 + lane) * 32;  // this lane's 16 v2f

        const v4f es0 = *(const v4f*)(ldsEsq + cb + 8 * hi);
        const v4f es1 = *(const v4f*)(ldsEsq + cb + 8 * hi + 4);
        v8f acc = {es0.x, es0.y, es0.z, es0.w, es1.x, es1.y, es1.z, es1.w};

        #pragma unroll
        for (int q = 0; q < 8; ++q) {
            const v4f av = *(const v4f*)(ablk + q * 4);    // 2 fragments per b128
            v2f a0; a0.x = av.x; a0.y = av.y;
            v2f a1; a1.x = av.z; a1.y = av.w;
            acc = __builtin_amdgcn_wmma_f32_16x16x4_f32(
                false, a0, false, bfrag[2 * q], (short)0, acc, false, false);
            acc = __builtin_amdgcn_wmma_f32_16x16x4_f32(
                false, a1, false, bfrag[2 * q + 1], (short)0, acc, false, false);
        }

        #pragma unroll
        for (int r = 0; r < 8; ++r) {
            const int code = cb + 8 * hi + r;
            if (acc[r] > best) { best = acc[r]; bestIdx = code; }  // lowest code wins ties
        }
    }

    // ---- Merge lane pairs (L, L^16) sharing a row ----
    const float obest = __shfl_xor(best, 16, 32);
    const int   oidx  = __shfl_xor(bestIdx, 16, 32);
    if (obest > best || (obest == best && oidx < bestIdx)) {
        best = obest; bestIdx = oidx;
    }
    if (hi == 0) ldsCode[wave * ROWS_PER_WAVE + lo] = bestIdx;
    __syncthreads();

    // ---- Gather quantized rows from swizzled codebook ----
    #pragma unroll 1
    for (int r = 0; r < ROWS_PER_WAVE; ++r) {
        const int code = ldsCode[wave * ROWS_PER_WAVE + r];
        const int ct = code >> 4;
        const int l2 = code & 15;
        const float* blk0 = ldsA + (ct * 32 + l2) * 32;
        const float* blk1 = ldsA + (ct * 32 + l2 + 16) * 32;
        float* orow = out + (size_t)(rowbase + r) * DDIM;
        #pragma unroll
        for (int t = 0; t < 2; ++t) {
            const int dd  = lane + 32 * t;
            const int j   = dd >> 2;
            const int rem = dd & 3;
            const float* blk = (rem & 2) ? blk1 : blk0;
#if USE_ASYNC_LDS
            // LDS -> global directly on the async engine (no VGPR round trip).
            float* lsrc = const_cast<float*>(blk + j * 2 + (rem & 1));
            __builtin_amdgcn_global_store_async_from_lds_b32(
                (gint_t)(orow + dd), (lint_t)lsrc, 0, 0);
#else
            orow[dd] = blk[j * 2 + (rem & 1)];
#endif
        }
    }
#if USE_ASYNC_LDS
    __builtin_amdgcn_s_wait_asynccnt(0);
#endif
}

extern "C" void kernel_launch(void* const* d_in, const int* in_sizes, int n_in,
                              void* d_out, int out_size, void* d_ws, size_t ws_size,
                              hipStream_t stream) {
    const float* x   = (const float*)d_in[0];   // [B,H,W,D] f32, flat N x 64
    const float* emb = (const float*)d_in[1];   // [D,K] f32
    float* out = (float*)d_out;

    const int N = in_sizes[0] / DDIM;           // 131072
    const int grid = N / ROWS_PER_WG;           // 1024 workgroups

    (void)hipFuncSetAttribute((const void*)vq_argmin_gather_kernel,
                              hipFuncAttributeMaxDynamicSharedMemorySize, LDS_BYTES);

    vq_argmin_gather_kernel<<<grid, THREADS, LDS_BYTES, stream>>>(x, emb, out);
}